// RNA_feature_extraction_23373212025451
// MI455X (gfx1250) — compile-verified
//
#include <hip/hip_runtime.h>

#define H 128

typedef __attribute__((ext_vector_type(2))) float v2f;
typedef __attribute__((ext_vector_type(4))) float v4f;
typedef __attribute__((ext_vector_type(8))) float v8f;

__device__ __forceinline__ void atomic_add_f32(float* p, float v) {
  __hip_atomic_fetch_add(p, v, __ATOMIC_RELAXED, __HIP_MEMORY_SCOPE_AGENT);
}

// ---------------------------------------------------------------------------
// Fold the rank-1 edge encoder through the per-layer edge Linear:
//   u[k] = sum_i ee_w[i] * lin_w[i,k] ;  v[k] = sum_i ee_b[i] * lin_w[i,k] + lin_b[k]
// ---------------------------------------------------------------------------
__global__ void prep_kernel(const float* __restrict__ ee_w, const float* __restrict__ ee_b,
                            const float* __restrict__ lin1_w, const float* __restrict__ lin1_b,
                            const float* __restrict__ lin2_w, const float* __restrict__ lin2_b,
                            float* __restrict__ u1, float* __restrict__ v1,
                            float* __restrict__ u2, float* __restrict__ v2) {
  int k = threadIdx.x;  // 128 threads
  float s1 = 0.f, t1 = 0.f, s2 = 0.f, t2 = 0.f;
  for (int i = 0; i < H; ++i) {
    float w1 = lin1_w[i * H + k];
    float w2 = lin2_w[i * H + k];
    float ew = ee_w[i], eb = ee_b[i];
    s1 += ew * w1; t1 += eb * w1;
    s2 += ew * w2; t2 += eb * w2;
  }
  u1[k] = s1; v1[k] = t1 + lin1_b[k];
  u2[k] = s2; v2[k] = t2 + lin2_b[k];
}

__global__ void zero_kernel(float* __restrict__ p, long n) {
  long i = (long)blockIdx.x * blockDim.x + threadIdx.x;
  long stride = (long)gridDim.x * blockDim.x;
  for (; i < n; i += stride) p[i] = 0.f;
}

// ---------------------------------------------------------------------------
// Layer-1 message pass: m = relu(x[src]*ne_w + ne_b + ea*u1 + v1), scatter-add
// One wave32 per edge, 4 columns per lane.
// ---------------------------------------------------------------------------
__global__ __launch_bounds__(256) void msg1_kernel(
    const float* __restrict__ x, const int* __restrict__ src, const int* __restrict__ dst,
    const float* __restrict__ ea, const float* __restrict__ ne_w, const float* __restrict__ ne_b,
    const float* __restrict__ u1, const float* __restrict__ v1,
    float* __restrict__ agg, long E) {
  int wave = threadIdx.x >> 5;
  int lane = threadIdx.x & 31;
  long e = (long)blockIdx.x * 8 + wave;
  if (e >= E) return;
  int s = src[e], d = dst[e];
  float xs = x[s];
  float av = ea[e];
  int k0 = lane * 4;
  v4f nw = *(const v4f*)(ne_w + k0);
  v4f nb = *(const v4f*)(ne_b + k0);
  v4f uu = *(const v4f*)(u1 + k0);
  v4f vv = *(const v4f*)(v1 + k0);
  float* out = agg + (long)d * H + k0;
#pragma unroll
  for (int j = 0; j < 4; ++j) {
    float m = xs * nw[j] + nb[j] + av * uu[j] + vv[j];
    m = m > 0.f ? m : 0.f;
    atomic_add_f32(out + j, m);
  }
}

// ---------------------------------------------------------------------------
// Layer-2 message pass: m = relu(h1[src] + ea*u2 + v2), scatter-add
// ---------------------------------------------------------------------------
__global__ __launch_bounds__(256) void msg2_kernel(
    const float* __restrict__ hprev, const int* __restrict__ src, const int* __restrict__ dst,
    const float* __restrict__ ea, const float* __restrict__ u2, const float* __restrict__ v2,
    float* __restrict__ agg, long E) {
  int wave = threadIdx.x >> 5;
  int lane = threadIdx.x & 31;
  long e = (long)blockIdx.x * 8 + wave;
  if (e >= E) return;
  int s = src[e], d = dst[e];
  float av = ea[e];
  int k0 = lane * 4;
  v4f hv = *(const v4f*)(hprev + (long)s * H + k0);
  v4f uu = *(const v4f*)(u2 + k0);
  v4f vv = *(const v4f*)(v2 + k0);
  float* out = agg + (long)d * H + k0;
#pragma unroll
  for (int j = 0; j < 4; ++j) {
    float m = hv[j] + av * uu[j] + vv[j];
    m = m > 0.f ? m : 0.f;
    atomic_add_f32(out + j, m);
  }
}

// ---------------------------------------------------------------------------
// Fused node MLP:  out = relu( relu(Z @ W1 + b1) @ W2 + b2 )
//   FIRST: Z = x[n]*ne_w + ne_b + agg  (rank-1 encoder + aggregation)
//   else : Z = hprev + agg
//   POOL : scatter-add the result into per-graph pool sums instead of storing
// 256 threads = 8 waves; block computes 16 rows; wave owns one 16-col tile.
// Exact fp32 via V_WMMA_F32_16X16X4_F32 (K-loop of 32 per GEMM).
// ---------------------------------------------------------------------------
template <bool FIRST, bool POOL>
__global__ __launch_bounds__(256) void mlp_kernel(
    const float* __restrict__ x, const float* __restrict__ ne_w, const float* __restrict__ ne_b,
    const float* __restrict__ hprev, const float* __restrict__ agg,
    const float* __restrict__ W1, const float* __restrict__ B1,
    const float* __restrict__ W2, const float* __restrict__ B2,
    float* __restrict__ hout, const int* __restrict__ batch, float* __restrict__ pool_sums,
    int N) {
  __shared__ float zA[16 * H];
  __shared__ float z1[16 * H];
  const int tid = threadIdx.x;
  const int rowBase = blockIdx.x * 16;

  // Cooperative Z-tile load: thread t -> row t/16, 8 consecutive cols.
  {
    int r = tid >> 4;
    int c0 = (tid & 15) * 8;
    long gr = rowBase + r;
    if (gr > N - 1) gr = N - 1;  // clamp (keeps EXEC all-1 later)
    const float* ag = agg + gr * H;
#pragma unroll
    for (int j = 0; j < 8; ++j) {
      int c = c0 + j;
      float z;
      if (FIRST) z = x[gr] * ne_w[c] + ne_b[c] + ag[c];
      else       z = hprev[gr * H + c] + ag[c];
      zA[r * H + c] = z;
    }
  }
  __syncthreads();

  const int wave = tid >> 5, lane = tid & 31;
  const int half = lane >> 4, lm = lane & 15;
  const int colBase = wave * 16;

  // GEMM1: D1 = Z(16x128) @ W1(128x128) tile, + bias, relu -> z1 LDS
  v8f c1 = {};
  for (int kk = 0; kk < H; kk += 4) {
    v2f a, b;
    a.x = zA[lm * H + kk + half * 2 + 0];
    a.y = zA[lm * H + kk + half * 2 + 1];
    b.x = W1[(kk + half * 2 + 0) * H + colBase + lm];
    b.y = W1[(kk + half * 2 + 1) * H + colBase + lm];
    c1 = __builtin_amdgcn_wmma_f32_16x16x4_f32(false, a, false, b, (short)0, c1, false, false);
  }
  {
    float bias = B1[colBase + lm];
#pragma unroll
    for (int v = 0; v < 8; ++v) {
      float val = c1[v] + bias;
      val = val > 0.f ? val : 0.f;
      z1[(half * 8 + v) * H + colBase + lm] = val;
    }
  }
  __syncthreads();

  // GEMM2: D2 = Z1(16x128) @ W2(128x128) tile, + bias, outer relu
  v8f c2 = {};
  for (int kk = 0; kk < H; kk += 4) {
    v2f a, b;
    a.x = z1[lm * H + kk + half * 2 + 0];
    a.y = z1[lm * H + kk + half * 2 + 1];
    b.x = W2[(kk + half * 2 + 0) * H + colBase + lm];
    b.y = W2[(kk + half * 2 + 1) * H + colBase + lm];
    c2 = __builtin_amdgcn_wmma_f32_16x16x4_f32(false, a, false, b, (short)0, c2, false, false);
  }
  {
    float bias = B2[colBase + lm];
#pragma unroll
    for (int v = 0; v < 8; ++v) {
      float val = c2[v] + bias;
      val = val > 0.f ? val : 0.f;
      int row = rowBase + half * 8 + v;
      int col = colBase + lm;
      if (row < N) {
        if (POOL) {
          int g = batch[row];
          atomic_add_f32(&pool_sums[g * H + col], val);
        } else {
          hout[(long)row * H + col] = val;
        }
      }
    }
  }
}

__global__ void count_kernel(const int* __restrict__ batch, float* __restrict__ cnt, int N) {
  int i = blockIdx.x * blockDim.x + threadIdx.x;
  if (i < N) atomic_add_f32(&cnt[batch[i]], 1.0f);
}

__global__ void finalize_kernel(const float* __restrict__ sums, const float* __restrict__ cnt,
                                float* __restrict__ out, int total) {
  int i = blockIdx.x * blockDim.x + threadIdx.x;
  if (i < total) {
    float c = cnt[i / H];
    out[i] = sums[i] / fmaxf(c, 1.0f);
  }
}

// ---------------------------------------------------------------------------
extern "C" void kernel_launch(void* const* d_in, const int* in_sizes, int n_in,
                              void* d_out, int out_size, void* d_ws, size_t ws_size,
                              hipStream_t stream) {
  const float* x      = (const float*)d_in[0];
  const int*   eidx   = (const int*)d_in[1];
  const float* eattr  = (const float*)d_in[2];
  const int*   batch  = (const int*)d_in[3];
  const float* ne_w   = (const float*)d_in[4];
  const float* ne_b   = (const float*)d_in[5];
  const float* ee_w   = (const float*)d_in[6];
  const float* ee_b   = (const float*)d_in[7];
  const float* lin1_w = (const float*)d_in[8];
  const float* lin1_b = (const float*)d_in[9];
  const float* w11    = (const float*)d_in[10];
  const float* b11    = (const float*)d_in[11];
  const float* w12    = (const float*)d_in[12];
  const float* b12    = (const float*)d_in[13];
  const float* lin2_w = (const float*)d_in[14];
  const float* lin2_b = (const float*)d_in[15];
  const float* w21    = (const float*)d_in[16];
  const float* b21    = (const float*)d_in[17];
  const float* w22    = (const float*)d_in[18];
  const float* b22    = (const float*)d_in[19];

  const int  N = in_sizes[0];
  const long E = (long)in_sizes[1] / 2;
  const int  G = out_size / H;
  const int* src = eidx;
  const int* dst = eidx + E;

  // Workspace layout (256B aligned)
  char* ws = (char*)d_ws;
  size_t off = 0;
  auto alloc = [&](size_t bytes) { char* p = ws + off; off = (off + bytes + 255) & ~(size_t)255; return p; };
  float* agg   = (float*)alloc((size_t)N * H * sizeof(float));  // reused for both layers
  float* h1    = (float*)alloc((size_t)N * H * sizeof(float));
  float* u1    = (float*)alloc(H * sizeof(float));
  float* v1    = (float*)alloc(H * sizeof(float));
  float* u2    = (float*)alloc(H * sizeof(float));
  float* v2    = (float*)alloc(H * sizeof(float));
  float* psum  = (float*)alloc((size_t)G * H * sizeof(float));
  float* pcnt  = (float*)alloc((size_t)G * sizeof(float));
  (void)ws_size;

  const long NH = (long)N * H;
  const int ZB = 1024;                      // blocks for zeroing
  const int msgBlocks = (int)((E + 7) / 8); // 8 edges (waves) per block
  const int mlpBlocks = (N + 15) / 16;

  prep_kernel<<<1, H, 0, stream>>>(ee_w, ee_b, lin1_w, lin1_b, lin2_w, lin2_b, u1, v1, u2, v2);

  // ---- Layer 1 ----
  zero_kernel<<<ZB, 256, 0, stream>>>(agg, NH);
  msg1_kernel<<<msgBlocks, 256, 0, stream>>>(x, src, dst, eattr, ne_w, ne_b, u1, v1, agg, E);
  mlp_kernel<true, false><<<mlpBlocks, 256, 0, stream>>>(
      x, ne_w, ne_b, /*hprev*/nullptr, agg, w11, b11, w12, b12, h1, nullptr, nullptr, N);

  // ---- Layer 2 ----
  zero_kernel<<<ZB, 256, 0, stream>>>(agg, NH);
  zero_kernel<<<1, 256, 0, stream>>>(psum, (long)G * H);
  zero_kernel<<<1, 64, 0, stream>>>(pcnt, G);
  msg2_kernel<<<msgBlocks, 256, 0, stream>>>(h1, src, dst, eattr, u2, v2, agg, E);
  mlp_kernel<false, true><<<mlpBlocks, 256, 0, stream>>>(
      nullptr, nullptr, nullptr, h1, agg, w21, b21, w22, b22, nullptr, batch, psum, N);

  // ---- Mean pool ----
  count_kernel<<<(N + 255) / 256, 256, 0, stream>>>(batch, pcnt, N);
  finalize_kernel<<<(G * H + 255) / 256, 256, 0, stream>>>(psum, pcnt, (float*)d_out, G * H);
}